// Attention_22935125360900
// MI455X (gfx1250) — compile-verified
//
#include <hip/hip_runtime.h>

// ---------------------------------------------------------------------------
// Location-sensitive attention, B=32, T=4096, H=512, conv K=11.
//
// Memory-bound (0.5 FLOP/byte): floor = 2 passes over eh (537 MB) / 23.3 TB/s
// ~= 23 us. We keep fp32 exact and use V_WMMA_F32_16X16X4_F32 as the MAC
// engine for both GEMVs (matrix units are otherwise idle; redundant columns
// cost no extra bytes).
//
// Outputs (concatenated): sx (B*H floats) then a (B*T floats).
// Workspace: pax (B*T floats = 512 KB).
// ---------------------------------------------------------------------------

typedef float v2f __attribute__((ext_vector_type(2)));
typedef float v8f __attribute__((ext_vector_type(8)));

#define ATT_B 32
#define ATT_T 4096
#define ATT_H 512
#define ATT_K 11
#define ATT_PAD 5

// ---------------------------------------------------------------------------
// Kernel 1: pax[b,t] = dot(eh[b,t,:], dhx[b,:])
// One wave per 16-row t-tile. WMMA f32 16x16x4, K walks h in steps of 4.
//   A[m,k] = eh[b, t0+m, h0+k]   (16x4)  -> lane = m + 16*(k>=2), b64 load
//   B[k,n] = dhx[b, h0+k]  (all n equal) -> VGPR0/1 = dhx[h0+kk], dhx[h0+kk+1]
//   D[m,n] = pax[t0+m] for every n. Column 0 lives in lane 0 (M=0..7) and
//   lane 16 (M=8..15), 8 accumulator VGPRs each.
// ---------------------------------------------------------------------------
__global__ __launch_bounds__(256) void att_pax_kernel(
    const float* __restrict__ eh, const float* __restrict__ dhx,
    float* __restrict__ pax)
{
    const int lane = threadIdx.x & 31;
    const int wave = threadIdx.x >> 5;
    const int tile = blockIdx.x * 8 + wave;          // 0 .. B*T/16-1 (8192)
    const int b    = tile >> 8;                      // T/16 = 256 tiles per b
    const int t0   = (tile & 255) << 4;

    const int m  = lane & 15;                        // matrix row
    const int kk = (lane >> 4) << 1;                 // 0 or 2 (K sub-pair)

    const float* __restrict__ arow = eh + ((size_t)(b * ATT_T + t0 + m)) * ATT_H + kk;
    const float* __restrict__ bvec = dhx + (size_t)b * ATT_H + kk;

    v8f c = {0.f, 0.f, 0.f, 0.f, 0.f, 0.f, 0.f, 0.f};

    #pragma unroll 8
    for (int h0 = 0; h0 < ATT_H; h0 += 4) {
        v2f a  = *(const v2f*)(arow + h0);           // eh[t0+m, h0+kk .. +1]
        v2f bb = *(const v2f*)(bvec + h0);           // dhx[h0+kk .. +1]
        // (neg_a, A, neg_b, B, c_mod, C, reuse_a, reuse_b)
        c = __builtin_amdgcn_wmma_f32_16x16x4_f32(
                false, a, false, bb, (short)0, c, false, false);
    }

    float* __restrict__ prow = pax + (size_t)b * ATT_T + t0;
    if (lane == 0) {
        #pragma unroll
        for (int v = 0; v < 8; ++v) prow[v] = c[v];      // M = 0..7
    } else if (lane == 16) {
        #pragma unroll
        for (int v = 0; v < 8; ++v) prow[8 + v] = c[v];  // M = 8..15
    }
}

// ---------------------------------------------------------------------------
// Kernel 2: loc = conv1d_same(ax, w) + bias;  a = softmax(pax + loc, axis=T)
// One 256-thread block per batch row; 16 t-values per thread, kept in
// registers across the max/sum reductions (wave shuffle + LDS across waves).
// ---------------------------------------------------------------------------
__global__ __launch_bounds__(256) void att_conv_softmax_kernel(
    const float* __restrict__ pax, const float* __restrict__ ax,
    const float* __restrict__ conv_w, const float* __restrict__ conv_b,
    float* __restrict__ aout)
{
    __shared__ float red_max[8];
    __shared__ float red_sum[8];

    const int b    = blockIdx.x;
    const int tid  = threadIdx.x;
    const int lane = tid & 31;
    const int wave = tid >> 5;

    float w[ATT_K];
    #pragma unroll
    for (int j = 0; j < ATT_K; ++j) w[j] = conv_w[j];   // scalar (uniform) loads
    const float bias = conv_b[0];

    const float* __restrict__ axb = ax  + (size_t)b * ATT_T;
    const float* __restrict__ pb  = pax + (size_t)b * ATT_T;

    float p[16];
    float lmax = -__builtin_inff();
    #pragma unroll
    for (int i = 0; i < 16; ++i) {
        const int t = i * 256 + tid;                    // coalesced
        float acc = pb[t] + bias;
        #pragma unroll
        for (int j = 0; j < ATT_K; ++j) {
            const int ts = t + j - ATT_PAD;
            const float v = (ts >= 0 && ts < ATT_T) ? axb[ts] : 0.f;
            acc = fmaf(v, w[j], acc);
        }
        p[i] = acc;
        lmax = fmaxf(lmax, acc);
    }

    // block max
    #pragma unroll
    for (int off = 16; off; off >>= 1)
        lmax = fmaxf(lmax, __shfl_xor(lmax, off, 32));
    if (lane == 0) red_max[wave] = lmax;
    __syncthreads();
    float bmax = red_max[0];
    #pragma unroll
    for (int i = 1; i < 8; ++i) bmax = fmaxf(bmax, red_max[i]);

    // exp + block sum
    float lsum = 0.f;
    #pragma unroll
    for (int i = 0; i < 16; ++i) {
        const float e = __expf(p[i] - bmax);
        p[i] = e;
        lsum += e;
    }
    #pragma unroll
    for (int off = 16; off; off >>= 1)
        lsum += __shfl_xor(lsum, off, 32);
    if (lane == 0) red_sum[wave] = lsum;
    __syncthreads();
    float bsum = 0.f;
    #pragma unroll
    for (int i = 0; i < 8; ++i) bsum += red_sum[i];
    const float inv = 1.f / bsum;

    float* __restrict__ ab = aout + (size_t)b * ATT_T;
    #pragma unroll
    for (int i = 0; i < 16; ++i)
        ab[i * 256 + tid] = p[i] * inv;
}

// ---------------------------------------------------------------------------
// Kernel 3: sx[b,h] = sum_t eh[b,t,h] * a[b,t]
// WMMA with roles swapped:
//   A[m,k] = a[b, t+k]      (replicated over m) -> b64 load of a[t+kk .. +1]
//   B[k,n] = eh[b, t+k, h0+n]                   -> two strided b32 loads
//   D[m,n] = partial sx[h0+n] (all m equal); lane n<16 holds it in c[0].
// t is split 8 ways per (b, h-chunk); partials via native f32 global atomics
// into a memset-zeroed sx region.
// ---------------------------------------------------------------------------
__global__ __launch_bounds__(256) void att_sx_kernel(
    const float* __restrict__ eh, const float* __restrict__ att,
    float* __restrict__ sx)
{
    const int lane = threadIdx.x & 31;
    const int wave = threadIdx.x >> 5;
    const int wid  = blockIdx.x * 8 + wave;          // 0 .. 8191
    const int tsl  = wid & 7;                        // t slice (8 x 512)
    const int hck  = (wid >> 3) & 31;                // h chunk (32 x 16)
    const int b    = wid >> 8;

    const int h0 = hck << 4;
    const int tb = tsl << 9;                         // 512 t per slice
    const int m  = lane & 15;                        // also the N column index
    const int kk = (lane >> 4) << 1;                 // 0 or 2

    const float* __restrict__ av  = att + (size_t)b * ATT_T + tb + kk;
    const float* __restrict__ ehb = eh + ((size_t)(b * ATT_T + tb)) * ATT_H + h0 + m;

    v8f c = {0.f, 0.f, 0.f, 0.f, 0.f, 0.f, 0.f, 0.f};

    #pragma unroll 4
    for (int dt = 0; dt < 512; dt += 4) {
        v2f a;                                        // A: att weights
        a.x = av[dt];
        a.y = av[dt + 1];
        const float* pB = ehb + (size_t)(dt + kk) * ATT_H;
        v2f bb;                                       // B: eh rows t+kk, t+kk+1
        bb.x = pB[0];
        bb.y = pB[ATT_H];
        c = __builtin_amdgcn_wmma_f32_16x16x4_f32(
                false, a, false, bb, (short)0, c, false, false);
    }

    if (lane < 16) {
        // c[0] = D[0, lane] = partial sx[h0+lane]
        __hip_atomic_fetch_add(&sx[(size_t)b * ATT_H + h0 + lane], c[0],
                               __ATOMIC_RELAXED, __HIP_MEMORY_SCOPE_AGENT);
    }
}

// ---------------------------------------------------------------------------
extern "C" void kernel_launch(void* const* d_in, const int* in_sizes, int n_in,
                              void* d_out, int out_size, void* d_ws, size_t ws_size,
                              hipStream_t stream) {
    (void)in_sizes; (void)n_in; (void)out_size; (void)ws_size;

    const float* eh  = (const float*)d_in[0];  // (B,T,H)
    const float* dhx = (const float*)d_in[1];  // (B,H)
    const float* ax  = (const float*)d_in[2];  // (B,T)
    const float* cw  = (const float*)d_in[3];  // (1,1,K)
    const float* cb  = (const float*)d_in[4];  // (1,)

    float* out  = (float*)d_out;
    float* sx   = out;                         // B*H floats
    float* aout = out + ATT_B * ATT_H;         // B*T floats
    float* pax  = (float*)d_ws;                // B*T floats scratch

    // zero sx accumulator region (graph-capturable memset node)
    hipMemsetAsync(sx, 0, (size_t)ATT_B * ATT_H * sizeof(float), stream);

    // 1) content scores: 8192 waves, 8 waves/block
    att_pax_kernel<<<ATT_B * (ATT_T / 16) / 8, 256, 0, stream>>>(eh, dhx, pax);

    // 2) conv + softmax over T, one block per batch row
    att_conv_softmax_kernel<<<ATT_B, 256, 0, stream>>>(pax, ax, cw, cb, aout);

    // 3) weighted summary: 8192 waves, atomics into zeroed sx
    att_sx_kernel<<<ATT_B * (ATT_H / 16) * 8 / 8, 256, 0, stream>>>(eh, aout, sx);
}